// RPNProposal_21784074125319
// MI455X (gfx1250) — compile-verified
//
#include <hip/hip_runtime.h>
#include <stdint.h>

#define N_ANCHORS  2000000
#define PRE        6000
#define POST       2000
#define PRE_PAD    6016              // 376 * 16
#define TILES      (PRE_PAD / 16)    // 376 tiles of 16
#define MASK_WORDS (PRE_PAD / 32)    // 188 u32 words per row
#define NB         2048              // histogram bins
#define CAND_CAP   16384             // candidate cap (power of two for bitonic)
#define IOU_THR_F  0.7f

typedef __attribute__((ext_vector_type(2))) float v2f;
typedef __attribute__((ext_vector_type(8))) float v8f;

// ---------------- workspace layout (bytes) ----------------
// hist        : int[NB]                     @ 0
// counter     : int                         @ NB*4
// thrBin      : int                         @ NB*4 + 4
// cand (u64)  : u64[CAND_CAP]               @ 8208
// top_score   : float[PRE_PAD]              @ 139280
// top_idx     : int[PRE_PAD]                @ 163344
// boxes       : float[PRE_PAD*4]            @ 187408
// mask16      : u16[PRE_PAD*TILES]          @ 283664   (~4.52 MB)
// total ~4.81 MB

// ---------------------------------------------------------------------------
__global__ void k_init(int* __restrict__ hist, int* __restrict__ counter) {
    int t = blockIdx.x * blockDim.x + threadIdx.x;
    if (t < NB) hist[t] = 0;
    if (t == 0) counter[0] = 0;
}

// ---------------------------------------------------------------------------
__global__ void __launch_bounds__(256)
k_hist(const float* __restrict__ scores, int* __restrict__ hist) {
    __shared__ int lhist[NB];
    for (int b = threadIdx.x; b < NB; b += blockDim.x) lhist[b] = 0;
    __syncthreads();
    int stride = gridDim.x * blockDim.x;
    for (int i = blockIdx.x * blockDim.x + threadIdx.x; i < N_ANCHORS; i += stride) {
        __builtin_prefetch(scores + i + 8 * stride, 0, 0);  // global_prefetch_b8
        float s = scores[i];
        int b = (int)(s * (float)NB);
        b = b < 0 ? 0 : (b > NB - 1 ? NB - 1 : b);
        atomicAdd(&lhist[b], 1);
    }
    __syncthreads();
    for (int b = threadIdx.x; b < NB; b += blockDim.x) {
        int v = lhist[b];
        if (v) atomicAdd(&hist[b], v);
    }
}

// ---------------------------------------------------------------------------
__global__ void k_thresh(const int* __restrict__ hist, int* __restrict__ thrBin) {
    if (threadIdx.x == 0 && blockIdx.x == 0) {
        int acc = 0, b = 0;
        for (int k = NB - 1; k >= 0; --k) {
            acc += hist[k];
            if (acc >= PRE) { b = k; break; }
        }
        thrBin[0] = b;
    }
}

// ---------------------------------------------------------------------------
__global__ void __launch_bounds__(256)
k_compact(const float* __restrict__ scores, const int* __restrict__ thrBin,
          int* __restrict__ counter, unsigned long long* __restrict__ cand) {
    float thr = (float)thrBin[0] * (1.0f / (float)NB);  // lower edge of threshold bin
    int stride = gridDim.x * blockDim.x;
    for (int i = blockIdx.x * blockDim.x + threadIdx.x; i < N_ANCHORS; i += stride) {
        __builtin_prefetch(scores + i + 8 * stride, 0, 0);
        float s = scores[i];
        if (s >= thr) {
            int pos = atomicAdd(counter, 1);
            if (pos < CAND_CAP) {
                unsigned int u = __float_as_uint(s);  // scores >= 0 -> monotone bits
                // descending by score, ascending by index on ties (matches top_k)
                unsigned long long key =
                    ((unsigned long long)(~u) << 32) | (unsigned int)i;
                cand[pos] = key;
            }
        }
    }
}

// ---------------------------------------------------------------------------
// Single-workgroup bitonic sort of up to 16K u64 keys, entirely in LDS.
// 128 KB dynamic LDS -- legal only because CDNA5 has 320 KB per WGP.
__global__ void __launch_bounds__(1024)
k_sort(const unsigned long long* __restrict__ cand, const int* __restrict__ counter,
       float* __restrict__ top_score, int* __restrict__ top_idx) {
    extern __shared__ unsigned long long s_keys[];
    int cnt = counter[0];
    if (cnt > CAND_CAP) cnt = CAND_CAP;
    int t = threadIdx.x;
    for (int e = t; e < CAND_CAP; e += blockDim.x)
        s_keys[e] = (e < cnt) ? cand[e] : ~0ULL;
    __syncthreads();
    for (int k = 2; k <= CAND_CAP; k <<= 1) {
        for (int j = k >> 1; j > 0; j >>= 1) {
            for (int e = t; e < CAND_CAP; e += blockDim.x) {
                int p = e ^ j;
                if (p > e) {
                    bool asc = ((e & k) == 0);
                    unsigned long long a = s_keys[e], b = s_keys[p];
                    if ((a > b) == asc) { s_keys[e] = b; s_keys[p] = a; }
                }
            }
            __syncthreads();
        }
    }
    for (int e = t; e < PRE_PAD; e += blockDim.x) {
        if (e < PRE) {
            unsigned long long key = s_keys[e];
            unsigned int hi = (unsigned int)(key >> 32);
            int idx = (int)(unsigned int)key;
            if (idx < 0 || idx >= N_ANCHORS) { top_score[e] = 0.0f; top_idx[e] = -1; }
            else { top_score[e] = __uint_as_float(~hi); top_idx[e] = idx; }
        } else {
            top_score[e] = 0.0f; top_idx[e] = -1;
        }
    }
}

// ---------------------------------------------------------------------------
// Decode only the surviving 6000 anchors (everything else is wasted bandwidth).
__global__ void __launch_bounds__(256)
k_decode(const float* __restrict__ enc, const float* __restrict__ anc,
         const int* __restrict__ top_idx, float* __restrict__ boxes) {
    int t = blockIdx.x * blockDim.x + threadIdx.x;
    if (t >= PRE_PAD) return;
    int i = (t < PRE) ? top_idx[t] : -1;
    float b0 = 0.f, b1 = 0.f, b2 = 0.f, b3 = 0.f;
    if (i >= 0) {
        float a0 = anc[(size_t)i * 4 + 0], a1 = anc[(size_t)i * 4 + 1];
        float a2 = anc[(size_t)i * 4 + 2], a3 = anc[(size_t)i * 4 + 3];
        float ha = a2 - a0, wa = a3 - a1;
        float cya = a0 + 0.5f * ha, cxa = a1 + 0.5f * wa;
        float ty = enc[(size_t)i * 4 + 0], tx = enc[(size_t)i * 4 + 1];
        float th = enc[(size_t)i * 4 + 2], tw = enc[(size_t)i * 4 + 3];
        float cy = ty * ha + cya, cx = tx * wa + cxa;
        float h = expf(th) * ha, w = expf(tw) * wa;
        b0 = cy - 0.5f * h; b1 = cx - 0.5f * w;
        b2 = cy + 0.5f * h; b3 = cx + 0.5f * w;
    }
    boxes[(size_t)t * 4 + 0] = b0; boxes[(size_t)t * 4 + 1] = b1;
    boxes[(size_t)t * 4 + 2] = b2; boxes[(size_t)t * 4 + 3] = b3;
}

// ---------------------------------------------------------------------------
// IoU mask kernel: one wave per 16x16 tile pair. The union base
// (area_m + area_n) for the whole tile comes from ONE rank-2
// v_wmma_f32_16x16x4_f32 that co-executes with the VALU intersection math.
__global__ void __launch_bounds__(32)
k_mask(const float* __restrict__ boxes, unsigned short* __restrict__ mask16) {
    __shared__ float rb[16][4], cb[16][4], rArea[16], cArea[16];
    int ti = blockIdx.y, tj = blockIdx.x;
    int lane = threadIdx.x;
    int r0 = ti * 16, c0 = tj * 16;

    if (lane < 16) {
        int m = lane;
        rb[m][0] = boxes[(size_t)(r0 + m) * 4 + 0];
        rb[m][1] = boxes[(size_t)(r0 + m) * 4 + 1];
        rb[m][2] = boxes[(size_t)(r0 + m) * 4 + 2];
        rb[m][3] = boxes[(size_t)(r0 + m) * 4 + 3];
    } else {
        int n = lane - 16;
        cb[n][0] = boxes[(size_t)(c0 + n) * 4 + 0];
        cb[n][1] = boxes[(size_t)(c0 + n) * 4 + 1];
        cb[n][2] = boxes[(size_t)(c0 + n) * 4 + 2];
        cb[n][3] = boxes[(size_t)(c0 + n) * 4 + 3];
    }
    __syncthreads();
    if (lane < 16)
        rArea[lane] = (rb[lane][2] - rb[lane][0]) * (rb[lane][3] - rb[lane][1]);
    else
        cArea[lane - 16] = (cb[lane - 16][2] - cb[lane - 16][0]) *
                           (cb[lane - 16][3] - cb[lane - 16][1]);
    __syncthreads();

    // A (16x4): lanes 0-15 hold K=0,1; lanes 16-31 hold K=2,3 (zeroed).
    // A[m] = [area_m, 1, 0, 0]; B[:,n] = [1, area_n, 0, 0]^T
    // => C[m][n] = area_m + area_n (exact in fp32).
    v2f a, b;
    if (lane < 16) {
        a.x = rArea[lane]; a.y = 1.0f;
        b.x = 1.0f;        b.y = cArea[lane];
    } else {
        a.x = 0.0f; a.y = 0.0f; b.x = 0.0f; b.y = 0.0f;
    }
    v8f c = {};
    c = __builtin_amdgcn_wmma_f32_16x16x4_f32(false, a, false, b, (short)0, c,
                                              false, false);

    // C/D layout: lane n in [0,15] + VGPR v -> (row v, col n);
    //             lane 16+n + VGPR v -> (row v+8, col n).
    int n    = lane & 15;
    int half = lane >> 4;
    float ny1 = cb[n][0], nx1 = cb[n][1], ny2 = cb[n][2], nx2 = cb[n][3];
    for (int v = 0; v < 8; ++v) {
        int m = v + 8 * half;
        float my1 = rb[m][0], mx1 = rb[m][1], my2 = rb[m][2], mx2 = rb[m][3];
        float iy1 = fmaxf(my1, ny1), ix1 = fmaxf(mx1, nx1);
        float iy2 = fminf(my2, ny2), ix2 = fminf(mx2, nx2);
        float inter = fmaxf(iy2 - iy1, 0.0f) * fmaxf(ix2 - ix1, 0.0f);
        float uni = fmaxf(c[v] - inter, 1e-8f);
        // iou > thr  <=>  inter > thr * max(union, 1e-8)   (no division)
        bool over = (inter > IOU_THR_F * uni) && ((c0 + n) > (r0 + m));
        unsigned int bal = __builtin_amdgcn_ballot_w32(over);
        if (lane == 0) {  // bits[15:0] = row v, bits[31:16] = row v+8
            mask16[(size_t)(r0 + v) * TILES + tj]     = (unsigned short)(bal & 0xFFFFu);
            mask16[(size_t)(r0 + v + 8) * TILES + tj] = (unsigned short)(bal >> 16);
        }
    }
}

// ---------------------------------------------------------------------------
// Sequential greedy scan with LDS suppressed-bitset; early exit at 2000 keeps.
__global__ void __launch_bounds__(256)
k_nms(const float* __restrict__ boxes, const float* __restrict__ top_score,
      const unsigned short* __restrict__ mask16, float* __restrict__ out) {
    __shared__ unsigned int sup[MASK_WORDS];
    __shared__ int s_keep, s_kc;
    int t = threadIdx.x;
    for (int w = t; w < POST * 5; w += 256) out[w] = 0.0f;   // d_out is poisoned
    for (int w = t; w < MASK_WORDS; w += 256) sup[w] = 0u;
    if (t == 0) s_kc = 0;
    __syncthreads();
    for (int i = 0; i < PRE; ++i) {
        if (t == 0) s_keep = ((sup[i >> 5] >> (i & 31)) & 1u) ? 0 : 1;
        __syncthreads();
        if (s_keep) {
            const unsigned int* row =
                (const unsigned int*)(mask16 + (size_t)i * TILES);  // 752B/row, 4B aligned
            for (int w = t; w < MASK_WORDS; w += 256) sup[w] |= row[w];
            if (t == 0) {
                int kc = s_kc;
                if (kc < POST) {
                    out[kc * 4 + 0] = boxes[(size_t)i * 4 + 0];
                    out[kc * 4 + 1] = boxes[(size_t)i * 4 + 1];
                    out[kc * 4 + 2] = boxes[(size_t)i * 4 + 2];
                    out[kc * 4 + 3] = boxes[(size_t)i * 4 + 3];
                    out[POST * 4 + kc] = top_score[i];
                    s_kc = kc + 1;
                }
            }
        }
        __syncthreads();
        if (s_kc >= POST) break;
    }
}

// ---------------------------------------------------------------------------
extern "C" void kernel_launch(void* const* d_in, const int* in_sizes, int n_in,
                              void* d_out, int out_size, void* d_ws, size_t ws_size,
                              hipStream_t stream) {
    const float* enc    = (const float*)d_in[0];
    const float* anc    = (const float*)d_in[1];
    const float* scores = (const float*)d_in[2];
    float* out = (float*)d_out;

    char* ws = (char*)d_ws;
    int* hist                     = (int*)(ws + 0);
    int* counter                  = (int*)(ws + (size_t)NB * 4);
    int* thrBin                   = (int*)(ws + (size_t)NB * 4 + 4);
    unsigned long long* cand      = (unsigned long long*)(ws + 8208);
    float* top_score              = (float*)(ws + 139280);
    int* top_idx                  = (int*)(ws + 163344);
    float* boxes                  = (float*)(ws + 187408);
    unsigned short* mask16        = (unsigned short*)(ws + 283664);

    k_init<<<(NB + 255) / 256, 256, 0, stream>>>(hist, counter);
    k_hist<<<1024, 256, 0, stream>>>(scores, hist);
    k_thresh<<<1, 32, 0, stream>>>(hist, thrBin);
    k_compact<<<2048, 256, 0, stream>>>(scores, thrBin, counter, cand);
    k_sort<<<1, 1024, (size_t)CAND_CAP * 8, stream>>>(cand, counter, top_score, top_idx);
    k_decode<<<(PRE_PAD + 255) / 256, 256, 0, stream>>>(enc, anc, top_idx, boxes);
    k_mask<<<dim3(TILES, TILES), 32, 0, stream>>>(boxes, mask16);
    k_nms<<<1, 256, 0, stream>>>(boxes, top_score, mask16, out);
}